// PolyConvFrame_73924977099081
// MI455X (gfx1250) — compile-verified
//
#include <hip/hip_runtime.h>
#include <stdint.h>
#include <math.h>

// Problem constants (fixed by the reference).
#define N_NODE  100000
#define DEPTH   10
#define N_EDGE  1600000
#define D_FEAT  32
#define STRIDE  ((DEPTH + 1) * D_FEAT)   // 352 floats per node in the output

// TDM staging tile: must divide N_EDGE exactly (1600000 / 2000 = 800 blocks).
#define TILE            2000
#define WAVES_PER_BLOCK 8

typedef unsigned int v4u __attribute__((ext_vector_type(4)));
typedef int          v8i __attribute__((ext_vector_type(8)));
typedef int          v4i __attribute__((ext_vector_type(4)));

// ---------------------------------------------------------------------------
// Tensor Data Mover: load a (d0 x d1) tile of 4-byte elements into LDS.
// Descriptor layout per CDNA5 ISA:
//   group0: count=1 | lds_addr | global_addr(57b) | type=2
//   group1: data_size=2(4B), tensor_dim0=d0, tensor_dim1=d1,
//           tile_dim0=d0, tile_dim1=d1, tensor_dim0_stride=s0 (elements)
// Tile == requested region (exact division), so no OOB handling is needed.
// ---------------------------------------------------------------------------
__device__ __forceinline__ void tdm_load_dw(uint32_t lds_off, const void* gptr,
                                            uint32_t d0, uint32_t d1, uint64_t s0)
{
  const uint64_t ga = (uint64_t)(uintptr_t)gptr;
  const v4u g0 = { 1u,                                   // count=1, no gather
                   lds_off,                              // LDS byte address
                   (uint32_t)ga,                         // global_addr[31:0]
                   (uint32_t)((ga >> 32) & 0x01FFFFFFu)  // global_addr[56:32]
                     | (2u << 30) };                     // type=2 ("image")
  const v8i g1 = {
    (int)(2u << 16),                                               // data_size=4B
    (int)((d0 & 0xFFFFu) << 16),                                   // tensor_dim0[15:0]
    (int)(((d0 >> 16) & 0xFFFFu) | ((d1 & 0xFFFFu) << 16)),        // td0[31:16]|td1[15:0]
    (int)(((d1 >> 16) & 0xFFFFu) | ((d0 & 0xFFFFu) << 16)),        // td1[31:16]|tile_dim0
    (int)(d1 & 0xFFFFu),                                           // tile_dim1 (tile_dim2=0)
    (int)(uint32_t)(s0 & 0xFFFFFFFFu),                             // dim0_stride[31:0]
    (int)(uint32_t)((s0 >> 32) & 0xFFFFu),                         // dim0_stride[47:32]
    0
  };
  const v4i z4 = { 0, 0, 0, 0 };
#if __clang_major__ >= 23
  const v8i z8 = { 0, 0, 0, 0, 0, 0, 0, 0 };
  __builtin_amdgcn_tensor_load_to_lds(g0, g1, z4, z4, z8, 0);
#else
  __builtin_amdgcn_tensor_load_to_lds(g0, g1, z4, z4, 0);
#endif
}

// ---------------------------------------------------------------------------
// Init: out[:,0,:] = x ; out[:,1..10,:] = 0 ; deg = 0.  Fully coalesced
// (32 consecutive threads = 32 features of one node).
// ---------------------------------------------------------------------------
__global__ __launch_bounds__(256)
void init_kernel(const float* __restrict__ x, float* __restrict__ out,
                 float* __restrict__ deg)
{
  int tid = blockIdx.x * blockDim.x + threadIdx.x;
  if (tid >= N_NODE * D_FEAT) return;
  int i = tid >> 5, f = tid & 31;
  size_t base = (size_t)i * STRIDE;
  out[base + f] = x[(size_t)i * D_FEAT + f];
#pragma unroll
  for (int L = 1; L <= DEPTH; ++L) out[base + (size_t)L * D_FEAT + f] = 0.0f;
  if (f == 0) deg[i] = 0.0f;
}

__global__ __launch_bounds__(256)
void degree_kernel(const int* __restrict__ ei, float* __restrict__ deg)
{
  int e = blockIdx.x * blockDim.x + threadIdx.x;
  if (e >= N_EDGE) return;
  (void)__hip_atomic_fetch_add(&deg[ei[e]], 1.0f,
                               __ATOMIC_RELAXED, __HIP_MEMORY_SCOPE_AGENT);
}

__global__ __launch_bounds__(256)
void dinv_kernel(float* __restrict__ deg_dinv)
{
  int i = blockIdx.x * blockDim.x + threadIdx.x;
  if (i >= N_NODE) return;
  float d = deg_dinv[i];
  if (d < 0.5f) d += 1.0f;           // isolated nodes: deg -> deg+1
  deg_dinv[i] = 1.0f / sqrtf(d);     // deg^-1/2
}

__global__ __launch_bounds__(256)
void val_kernel(const int* __restrict__ ei, const float* __restrict__ w,
                const float* __restrict__ dinv, float* __restrict__ val)
{
  int e = blockIdx.x * blockDim.x + threadIdx.x;
  if (e >= N_EDGE) return;
  val[e] = dinv[ei[e]] * w[e] * dinv[ei[N_EDGE + e]];
}

__global__ void alpha_kernel(const float* __restrict__ a, float* __restrict__ at)
{
  int t = threadIdx.x;
  if (t <= DEPTH) at[t] = tanhf(a[t]);   // BASE_ALPHA == 1.0
}

// ---------------------------------------------------------------------------
// One polynomial level:  hcur[row,:] += alpha * val[e] * hprev[col,:].
// TDM streams the row/col tiles (one 2D descriptor, dim0_stride = N_EDGE)
// and the val tile into LDS; compute waves then run lane-per-feature:
// coalesced 128B gather + coalesced no-return global_atomic_add_f32.
// ---------------------------------------------------------------------------
__global__ __launch_bounds__(256)
void spmv_level_kernel(const int* __restrict__ ei, const float* __restrict__ val,
                       const float* __restrict__ hprev, float* __restrict__ hcur,
                       const float* __restrict__ alpha_p)
{
  __shared__ int   s_ei[2 * TILE];   // [0,TILE) = row tile, [TILE,2*TILE) = col tile
  __shared__ float s_val[TILE];

  const int tile0 = blockIdx.x * TILE;
  if (threadIdx.x < 32) {            // one wave issues the DMAs (EXEC-independent op)
    tdm_load_dw((uint32_t)(uintptr_t)s_ei,  ei  + tile0, TILE, 2u, (uint64_t)N_EDGE);
    tdm_load_dw((uint32_t)(uintptr_t)s_val, val + tile0, TILE, 1u, 0);
    __builtin_amdgcn_s_wait_tensorcnt(0);
  }
  __syncthreads();

  const float alpha = *alpha_p;                 // uniform -> scalar load
  const int lane = threadIdx.x & 31;            // lane == feature (wave32)
  const int wave = threadIdx.x >> 5;
  const int per  = TILE / WAVES_PER_BLOCK;      // 250 edges per wave
  const int jend = (wave + 1) * per;
#pragma unroll 4
  for (int j = wave * per; j < jend; ++j) {
    const int   r = s_ei[j];
    const int   c = s_ei[TILE + j];
    const float v = s_val[j] * alpha;
    const float h = hprev[(size_t)c * STRIDE + lane];
    (void)__hip_atomic_fetch_add(hcur + (size_t)r * STRIDE + lane, v * h,
                                 __ATOMIC_RELAXED, __HIP_MEMORY_SCOPE_AGENT);
  }
}

// Fallback when ws is too small to hold val[]: stage edge_attr via TDM and
// fold the GCN normalization in per-edge (dinv reads broadcast from L2).
__global__ __launch_bounds__(256)
void spmv_level_kernel_nw(const int* __restrict__ ei, const float* __restrict__ w,
                          const float* __restrict__ dinv,
                          const float* __restrict__ hprev, float* __restrict__ hcur,
                          const float* __restrict__ alpha_p)
{
  __shared__ int   s_ei[2 * TILE];
  __shared__ float s_w[TILE];

  const int tile0 = blockIdx.x * TILE;
  if (threadIdx.x < 32) {
    tdm_load_dw((uint32_t)(uintptr_t)s_ei, ei + tile0, TILE, 2u, (uint64_t)N_EDGE);
    tdm_load_dw((uint32_t)(uintptr_t)s_w,  w  + tile0, TILE, 1u, 0);
    __builtin_amdgcn_s_wait_tensorcnt(0);
  }
  __syncthreads();

  const float alpha = *alpha_p;
  const int lane = threadIdx.x & 31;
  const int wave = threadIdx.x >> 5;
  const int per  = TILE / WAVES_PER_BLOCK;
  const int jend = (wave + 1) * per;
#pragma unroll 4
  for (int j = wave * per; j < jend; ++j) {
    const int   r = s_ei[j];
    const int   c = s_ei[TILE + j];
    const float v = dinv[r] * s_w[j] * dinv[c] * alpha;
    const float h = hprev[(size_t)c * STRIDE + lane];
    (void)__hip_atomic_fetch_add(hcur + (size_t)r * STRIDE + lane, v * h,
                                 __ATOMIC_RELAXED, __HIP_MEMORY_SCOPE_AGENT);
  }
}

extern "C" void kernel_launch(void* const* d_in, const int* in_sizes, int n_in,
                              void* d_out, int out_size, void* d_ws, size_t ws_size,
                              hipStream_t stream)
{
  (void)in_sizes; (void)n_in; (void)out_size;

  const float* x  = (const float*)d_in[0];   // (N_NODE, 32) f32
  const int*   ei = (const int*)d_in[1];     // (2, N_EDGE): rows then cols
  const float* w  = (const float*)d_in[2];   // (N_EDGE,) f32
  const float* al = (const float*)d_in[3];   // (DEPTH+1,) f32
  float* out = (float*)d_out;                // (N_NODE, DEPTH+1, 32) f32

  // Workspace layout: [deg/dinv: N][alphas_t: 16][val: E]
  float* deg = (float*)d_ws;
  float* at  = deg + N_NODE;
  float* val = at + 16;
  const bool have_val =
      ws_size >= (size_t)(N_NODE + 16 + N_EDGE) * sizeof(float);

  const dim3 b256(256);
  init_kernel  <<<dim3((N_NODE * D_FEAT + 255) / 256), b256, 0, stream>>>(x, out, deg);
  degree_kernel<<<dim3((N_EDGE + 255) / 256),          b256, 0, stream>>>(ei, deg);
  dinv_kernel  <<<dim3((N_NODE + 255) / 256),          b256, 0, stream>>>(deg);
  alpha_kernel <<<1, 32, 0, stream>>>(al, at);
  if (have_val)
    val_kernel <<<dim3((N_EDGE + 255) / 256),          b256, 0, stream>>>(ei, w, deg, val);

  const int nblk = N_EDGE / TILE;   // 800, exact
  for (int L = 1; L <= DEPTH; ++L) {
    const float* hprev = out + (size_t)(L - 1) * D_FEAT;   // slice L-1, stride 352
    float*       hcur  = out + (size_t)L * D_FEAT;         // slice L
    if (have_val)
      spmv_level_kernel   <<<nblk, b256, 0, stream>>>(ei, val, hprev, hcur, at + L);
    else
      spmv_level_kernel_nw<<<nblk, b256, 0, stream>>>(ei, w, deg, hprev, hcur, at + L);
  }
}